// MaskedTimeAttention_60988535603543
// MI455X (gfx1250) — compile-verified
//
#include <hip/hip_runtime.h>
#include <hip/hip_bf16.h>
#include <math.h>

// ---- problem dimensions (fixed by the reference) ----
#define NB   32
#define EN   512
#define NDE  64
#define DEN  128   // D_en
#define DDE  128   // D_de
#define NU   128   // U
#define REMOVE_FACTOR 10000.0f

typedef float v2f __attribute__((ext_vector_type(2)));
typedef float v8f __attribute__((ext_vector_type(8)));

__device__ __forceinline__ float fast_tanhf(float x) {
#if __has_builtin(__builtin_amdgcn_tanhf)
  return __builtin_amdgcn_tanhf(x);   // v_tanh_f32 (gfx1250 TRANS op)
#else
  return tanhf(x);
#endif
}

// One wave32 computes one 16x16 f32 tile of C = A(16xK) * B(Kx16)
// using V_WMMA_F32_16X16X4_F32, K stepped by 4.
// A pre-offset to row m0 (row-major, stride lda); B pre-offset to col n0
// (row-major, stride ldb).
// 32-bit A 16x4 layout: lane L -> row (L&15), VGPR0=K(2*(L>>4)), VGPR1=K+1.
// 32-bit B 4x16 layout (mirror):  lane L -> col (L&15), K likewise.
__device__ __forceinline__ v8f wmma_tile_f32(const float* __restrict__ A, int lda,
                                             const float* __restrict__ Bm, int ldb,
                                             int K, int lane) {
  v8f acc = {};
  const int half = lane >> 4;    // 0: K 0..1, 1: K 2..3
  const int idx  = lane & 15;    // row of A / col of B
  for (int k = 0; k < K; k += 4) {
    const int kk = k + 2 * half;
    v2f a = *(const v2f*)(A + idx * lda + kk);          // A[idx, kk..kk+1]
    v2f b;
    b.x = Bm[kk * ldb + idx];                           // B[kk,   idx]
    b.y = Bm[(kk + 1) * ldb + idx];                     // B[kk+1, idx]
    acc = __builtin_amdgcn_wmma_f32_16x16x4_f32(
        /*neg_a=*/false, a, /*neg_b=*/false, b,
        /*c_mod=*/(short)0, acc, /*reuse_a=*/false, /*reuse_b=*/false);
  }
  return acc;
}

// ---- 0) en_masked = en_seq * mask (broadcast over feature dim) ----
__global__ void k_mask_en(const float* __restrict__ en_seq,
                          const int* __restrict__ mask,
                          float* __restrict__ en_m) {
  int idx = blockIdx.x * blockDim.x + threadIdx.x;     // over NB*EN*DEN
  float m = (float)mask[idx / DEN];
  en_m[idx] = en_seq[idx] * m;
}

// ---- 1) att_enT[b,u,e] = (en_m[b] @ w_en)[e,u], stored u-major ----
__global__ void k_att_en(const float* __restrict__ en_m,
                         const float* __restrict__ w_en,
                         float* __restrict__ attT) {
  const int lane = threadIdx.x & 31;
  const int wid  = (blockIdx.x * blockDim.x + threadIdx.x) >> 5;
  const int MT = EN / 16, NT = NU / 16;                // 32 x 8 tiles per batch
  const int b  = wid / (MT * NT);
  const int t  = wid % (MT * NT);
  const int m0 = (t / NT) * 16;
  const int n0 = (t % NT) * 16;

  v8f acc = wmma_tile_f32(en_m + (size_t)b * EN * DEN + (size_t)m0 * DEN, DEN,
                          w_en + n0, NU, DEN, lane);

  // D layout: VGPR r, lane L -> (m = r + 8*(L>>4), n = L&15)
  const int half = lane >> 4, idx = lane & 15;
  float* out = attT + (size_t)b * NU * EN + (size_t)(n0 + idx) * EN + m0 + 8 * half;
#pragma unroll
  for (int r = 0; r < 8; ++r) out[r] = acc[r];
}

// ---- 2) att_de[b,d,u] = (de_seq[b] @ w_de)[d,u], row-major ----
__global__ void k_att_de(const float* __restrict__ de_seq,
                         const float* __restrict__ w_de,
                         float* __restrict__ att_de) {
  const int lane = threadIdx.x & 31;
  const int wid  = (blockIdx.x * blockDim.x + threadIdx.x) >> 5;
  const int MT = NDE / 16, NT = NU / 16;               // 4 x 8 tiles per batch
  const int b  = wid / (MT * NT);
  const int t  = wid % (MT * NT);
  const int m0 = (t / NT) * 16;
  const int n0 = (t % NT) * 16;

  v8f acc = wmma_tile_f32(de_seq + (size_t)b * NDE * DDE + (size_t)m0 * DDE, DDE,
                          w_de + n0, NU, DDE, lane);

  const int half = lane >> 4, idx = lane & 15;
  float* out = att_de + (size_t)b * NDE * NU;
#pragma unroll
  for (int r = 0; r < 8; ++r)
    out[(size_t)(m0 + r + 8 * half) * NU + n0 + idx] = acc[r];
}

// ---- 3) mu[b,d,e] = sum_u tanh(att_enT[b,u,e] + att_de[b,d,u]) * nu[u] ----
__global__ void k_mu(const float* __restrict__ attT,
                     const float* __restrict__ att_de,
                     const float* __restrict__ nu,
                     float* __restrict__ mu) {
  const int lane = threadIdx.x & 31;
  const int wid  = (blockIdx.x * blockDim.x + threadIdx.x) >> 5;
  const int ET = EN / 32;                              // 16 e-chunks per (b,d)
  const int b  = wid / (NDE * ET);
  const int r  = wid % (NDE * ET);
  const int d  = r / ET;
  const int e  = (r % ET) * 32 + lane;

  const float* ae = attT + (size_t)b * NU * EN + e;     // stride EN over u
  const float* ad = att_de + ((size_t)b * NDE + d) * NU;

  float acc = 0.f;
#pragma unroll 4
  for (int u = 0; u < NU; ++u)
    acc += fast_tanhf(ae[(size_t)u * EN] + ad[u]) * nu[u];

  mu[((size_t)b * NDE + d) * EN + e] = acc;
}

// ---- 4) alphas = softmax(mu + bias, axis=e); bias uses the reference's
//         flat-layout mask index: mask[b, d*8 + e/64] ----
__global__ void k_softmax(const int* __restrict__ mask,
                          float* __restrict__ mu) {
  const int lane = threadIdx.x & 31;
  const int wid  = (blockIdx.x * blockDim.x + threadIdx.x) >> 5; // over NB*NDE
  const int b = wid / NDE;
  const int d = wid % NDE;
  float* row = mu + (size_t)wid * EN;
  const int* mb = mask + (size_t)b * EN;

  float x[16];
  float mx = -3.0e38f;
#pragma unroll
  for (int i = 0; i < 16; ++i) {
    const int e = i * 32 + lane;
    const int mi = mb[(d * EN + e) >> 6];              // == d*8 + e/64
    const float bias = ((float)mi - 1.0f) * REMOVE_FACTOR;
    x[i] = row[e] + bias;
    mx = fmaxf(mx, x[i]);
  }
#pragma unroll
  for (int o = 16; o >= 1; o >>= 1) mx = fmaxf(mx, __shfl_xor(mx, o, 32));

  float s = 0.f;
#pragma unroll
  for (int i = 0; i < 16; ++i) { x[i] = __expf(x[i] - mx); s += x[i]; }
#pragma unroll
  for (int o = 16; o >= 1; o >>= 1) s += __shfl_xor(s, o, 32);

  const float inv = 1.0f / s;
#pragma unroll
  for (int i = 0; i < 16; ++i) row[i * 32 + lane] = x[i] * inv;
}

// ---- 5) out[b,d,128+f] = (alphas[b] @ en_m[b])[d,f]  (K = 512) ----
__global__ void k_sum_en(const float* __restrict__ alphas,
                         const float* __restrict__ en_m,
                         float* __restrict__ out) {
  const int lane = threadIdx.x & 31;
  const int wid  = (blockIdx.x * blockDim.x + threadIdx.x) >> 5;
  const int MT = NDE / 16, NT = DEN / 16;              // 4 x 8 tiles per batch
  const int b  = wid / (MT * NT);
  const int t  = wid % (MT * NT);
  const int m0 = (t / NT) * 16;
  const int n0 = (t % NT) * 16;

  v8f acc = wmma_tile_f32(alphas + (size_t)b * NDE * EN + (size_t)m0 * EN, EN,
                          en_m + (size_t)b * EN * DEN + n0, DEN, EN, lane);

  const int half = lane >> 4, idx = lane & 15;
  const int OW = DDE + DEN;                            // 256
  float* ob = out + (size_t)b * NDE * OW;
#pragma unroll
  for (int r = 0; r < 8; ++r)
    ob[(size_t)(m0 + r + 8 * half) * OW + DDE + n0 + idx] = acc[r];
}

// ---- 6) out[b,d,0:128] = de_seq[b,d,:] ----
__global__ void k_copy_de(const float* __restrict__ de_seq,
                          float* __restrict__ out) {
  int idx = blockIdx.x * blockDim.x + threadIdx.x;     // over NB*NDE*DDE
  int bd = idx / DDE;
  int f  = idx % DDE;
  out[(size_t)bd * (DDE + DEN) + f] = de_seq[idx];
}

extern "C" void kernel_launch(void* const* d_in, const int* in_sizes, int n_in,
                              void* d_out, int out_size, void* d_ws, size_t ws_size,
                              hipStream_t stream) {
  const float* en_seq = (const float*)d_in[0];   // (32,512,128)
  const float* de_seq = (const float*)d_in[1];   // (32,64,128)
  const int*   mask   = (const int*)d_in[2];     // (32,512)
  const float* w_en   = (const float*)d_in[3];   // (128,128)
  const float* w_de   = (const float*)d_in[4];   // (128,128)
  const float* nu     = (const float*)d_in[5];   // (128,1)
  float* out = (float*)d_out;                    // (32,64,256)

  // workspace layout (floats)
  float* ws      = (float*)d_ws;
  float* en_m    = ws;                                       // 2,097,152
  float* attT    = en_m + (size_t)NB * EN * DEN;             // 2,097,152
  float* att_de  = attT + (size_t)NB * NU * EN;              //   262,144
  float* mu      = att_de + (size_t)NB * NDE * NU;           // 1,048,576

  // 0) mask multiply: NB*EN*DEN = 2,097,152 threads
  k_mask_en<<<(NB * EN * DEN) / 256, 256, 0, stream>>>(en_seq, mask, en_m);

  // 1) att_enT: NB*(EN/16)*(NU/16) = 8192 waves, 4 waves/block
  k_att_en<<<8192 / 4, 128, 0, stream>>>(en_m, w_en, attT);

  // 2) att_de: NB*(NDE/16)*(NU/16) = 1024 waves
  k_att_de<<<1024 / 4, 128, 0, stream>>>(de_seq, w_de, att_de);

  // 3) mu: NB*NDE*(EN/32) = 32768 waves, 8 waves/block
  k_mu<<<32768 / 8, 256, 0, stream>>>(attT, att_de, nu, mu);

  // 4) softmax (in place on mu): NB*NDE = 2048 waves
  k_softmax<<<2048 / 8, 256, 0, stream>>>(mask, mu);

  // 5) sum_en -> out[...,128:256]: NB*(NDE/16)*(DEN/16) = 1024 waves
  k_sum_en<<<1024 / 4, 128, 0, stream>>>(mu, en_m, out);

  // 6) de_seq -> out[...,0:128]: NB*NDE*DDE = 262,144 threads
  k_copy_de<<<(NB * NDE * DDE) / 256, 256, 0, stream>>>(de_seq, out);
}